// Net_agnn_59768764892001
// MI455X (gfx1250) — compile-verified
//
#include <hip/hip_runtime.h>
#include <hip/hip_bf16.h>

#define NN    100000
#define EE    3200000
#define FIN   767
#define HH    16
#define CC    10
#define EPSN  1e-12f

typedef __attribute__((ext_vector_type(2))) float v2f;
typedef __attribute__((ext_vector_type(8))) float v8f;

// ---------------------------------------------------------------------------
// Kernel 1: h = relu(x @ W1^T + b1) via V_WMMA_F32_16X16X4_F32
// One wave computes a 16(node) x 16(hidden) tile, K-loop over 767 in steps of 4.
// A layout (16x4 f32): lane m=lane&15 holds row m; v0: K=k0+2*half, v1: +1.
// B layout (4x16 f32): lane n=lane&15 holds col n; v0: K=k0+2*half, v1: +1.
// C/D (16x16 f32): VGPR r = row r (lanes 0-15) / row r+8 (lanes 16-31).
// ---------------------------------------------------------------------------
__global__ __launch_bounds__(256) void gemm1_relu_kernel(
    const float* __restrict__ x, const float* __restrict__ w1,
    const float* __restrict__ b1, float* __restrict__ h)
{
    const int wave = (blockIdx.x * 256 + (int)threadIdx.x) >> 5;
    const int lane = (int)threadIdx.x & 31;
    if (wave >= NN / 16) return;          // wave-uniform: EXEC stays all-1s

    const int row0 = wave * 16;
    const int n    = lane & 15;           // A: row-in-tile (M); B: column (N)
    const int half = lane >> 4;

    const float* __restrict__ arow = x  + (size_t)(row0 + n) * FIN;
    const float* __restrict__ brow = w1 + (size_t)n * FIN;    // W1 row n == B col n

    v8f acc = {};
    int k0 = 0;
    #pragma unroll 4
    for (; k0 + 4 <= FIN; k0 += 4) {
        const int ka = k0 + 2 * half;
        v2f a;  a.x  = arow[ka]; a.y  = arow[ka + 1];
        v2f bb; bb.x = brow[ka]; bb.y = brow[ka + 1];
        acc = __builtin_amdgcn_wmma_f32_16x16x4_f32(
                  false, a, false, bb, (short)0, acc, false, false);
    }
    {   // K tail (767 = 4*191 + 3): zero-pad out-of-range K
        const int ka = k0 + 2 * half;
        v2f a;  a.x  = (ka     < FIN) ? arow[ka]     : 0.0f;
                a.y  = (ka + 1 < FIN) ? arow[ka + 1] : 0.0f;
        v2f bb; bb.x = (ka     < FIN) ? brow[ka]     : 0.0f;
                bb.y = (ka + 1 < FIN) ? brow[ka + 1] : 0.0f;
        acc = __builtin_amdgcn_wmma_f32_16x16x4_f32(
                  false, a, false, bb, (short)0, acc, false, false);
    }

    const float bias = b1[n];
    #pragma unroll
    for (int r = 0; r < 8; ++r) {
        float v = acc[r] + bias;
        v = v > 0.0f ? v : 0.0f;                       // ReLU
        h[(size_t)(row0 + r + 8 * half) * HH + n] = v; // half-wave writes one row
    }
}

// ---------------------------------------------------------------------------
// Kernel 2: row-wise L2 normalize: xn = v / max(||v||, eps)
// ---------------------------------------------------------------------------
__global__ __launch_bounds__(256) void normalize_kernel(
    const float* __restrict__ in, float* __restrict__ out)
{
    const int i = blockIdx.x * 256 + (int)threadIdx.x;
    if (i >= NN) return;
    const float4* __restrict__ r = (const float4*)(in + (size_t)i * HH);
    float4 a = r[0], b = r[1], c = r[2], d = r[3];
    float s = a.x*a.x + a.y*a.y + a.z*a.z + a.w*a.w
            + b.x*b.x + b.y*b.y + b.z*b.z + b.w*b.w
            + c.x*c.x + c.y*c.y + c.z*c.z + c.w*c.w
            + d.x*d.x + d.y*d.y + d.z*d.z + d.w*d.w;
    const float inv = 1.0f / fmaxf(sqrtf(s), EPSN);
    float4* __restrict__ o = (float4*)(out + (size_t)i * HH);
    a.x*=inv; a.y*=inv; a.z*=inv; a.w*=inv;
    b.x*=inv; b.y*=inv; b.z*=inv; b.w*=inv;
    c.x*=inv; c.y*=inv; c.z*=inv; c.w*=inv;
    d.x*=inv; d.y*=inv; d.z*=inv; d.w*=inv;
    o[0]=a; o[1]=b; o[2]=c; o[3]=d;
}

// ---------------------------------------------------------------------------
// Kernel 3: zero fill
// ---------------------------------------------------------------------------
__global__ __launch_bounds__(256) void zero_kernel(float* __restrict__ p, int n)
{
    const int i = blockIdx.x * 256 + (int)threadIdx.x;
    if (i < n) p[i] = 0.0f;
}

// ---------------------------------------------------------------------------
// Kernel 4: per-edge numerator. Softmax is shift-invariant and
// alpha = beta*cos in [-|beta|,|beta|] is bounded, so no segment-max needed:
//   ex_e = exp(beta * <xn[src], xn[dst]>);  denom[dst] += ex_e  (HW fp32 atomic)
// Edges [0,E) come from edge_index; edges [E, E+N) are the implicit self-loops.
// ---------------------------------------------------------------------------
__global__ __launch_bounds__(256) void edge_num_kernel(
    const int* __restrict__ ei, const float* __restrict__ xn,
    const float* __restrict__ beta_ptr, float* __restrict__ exbuf,
    float* __restrict__ denom)
{
    const int e = blockIdx.x * 256 + (int)threadIdx.x;
    if (e >= EE + NN) return;
    int s, d;
    if (e < EE) { s = ei[e]; d = ei[EE + e]; }
    else        { s = d = e - EE; }
    const float4* __restrict__ pa = (const float4*)(xn + (size_t)s * HH);
    const float4* __restrict__ pb = (const float4*)(xn + (size_t)d * HH);
    float dot = 0.0f;
    #pragma unroll
    for (int j = 0; j < 4; ++j) {
        float4 a = pa[j], b = pb[j];
        dot += a.x*b.x + a.y*b.y + a.z*b.z + a.w*b.w;
    }
    const float beta = beta_ptr ? beta_ptr[0] : 1.0f;
    const float ex = __expf(beta * dot);
    exbuf[e] = ex;
    unsafeAtomicAdd(denom + d, ex);       // global_atomic_add_f32
}

// ---------------------------------------------------------------------------
// Kernel 5: weighted scatter-aggregation: out[dst] += (ex/denom[dst]) * feat[src]
// ---------------------------------------------------------------------------
__global__ __launch_bounds__(256) void edge_agg_kernel(
    const int* __restrict__ ei, const float* __restrict__ exbuf,
    const float* __restrict__ denom, const float* __restrict__ feat,
    float* __restrict__ out)
{
    const int e = blockIdx.x * 256 + (int)threadIdx.x;
    if (e >= EE + NN) return;
    int s, d;
    if (e < EE) { s = ei[e]; d = ei[EE + e]; }
    else        { s = d = e - EE; }
    const float w = exbuf[e] / denom[d];
    const float4* __restrict__ f4 = (const float4*)(feat + (size_t)s * HH);
    float* __restrict__ o = out + (size_t)d * HH;
    #pragma unroll
    for (int j = 0; j < 4; ++j) {
        float4 v = f4[j];
        unsafeAtomicAdd(o + 4*j + 0, w * v.x);
        unsafeAtomicAdd(o + 4*j + 1, w * v.y);
        unsafeAtomicAdd(o + 4*j + 2, w * v.z);
        unsafeAtomicAdd(o + 4*j + 3, w * v.w);
    }
}

// ---------------------------------------------------------------------------
// Kernel 6: logits = h2 @ W2^T + b2 ; log_softmax over 10 classes
// W2 (10x16) + b2 staged in LDS.
// ---------------------------------------------------------------------------
__global__ __launch_bounds__(256) void head_kernel(
    const float* __restrict__ h2, const float* __restrict__ w2,
    const float* __restrict__ b2, float* __restrict__ out)
{
    __shared__ float sw[CC * HH + CC];
    if (threadIdx.x < CC * HH) sw[threadIdx.x] = w2[threadIdx.x];
    if (threadIdx.x < CC)      sw[CC * HH + threadIdx.x] = b2[threadIdx.x];
    __syncthreads();

    const int i = blockIdx.x * 256 + (int)threadIdx.x;
    if (i >= NN) return;

    float hv[HH];
    const float4* __restrict__ hr = (const float4*)(h2 + (size_t)i * HH);
    #pragma unroll
    for (int j = 0; j < 4; ++j) {
        float4 v = hr[j];
        hv[4*j+0]=v.x; hv[4*j+1]=v.y; hv[4*j+2]=v.z; hv[4*j+3]=v.w;
    }
    float l[CC];
    float m = -3.4e38f;
    #pragma unroll
    for (int c = 0; c < CC; ++c) {
        float acc = sw[CC * HH + c];
        #pragma unroll
        for (int f = 0; f < HH; ++f) acc += hv[f] * sw[c * HH + f];
        l[c] = acc;
        m = fmaxf(m, acc);
    }
    float se = 0.0f;
    #pragma unroll
    for (int c = 0; c < CC; ++c) se += __expf(l[c] - m);
    const float lse = m + __logf(se);
    float* __restrict__ o = out + (size_t)i * CC;
    #pragma unroll
    for (int c = 0; c < CC; ++c) o[c] = l[c] - lse;
}

// ---------------------------------------------------------------------------
extern "C" void kernel_launch(void* const* d_in, const int* in_sizes, int n_in,
                              void* d_out, int out_size, void* d_ws, size_t ws_size,
                              hipStream_t stream)
{
    const float* x     = (const float*)d_in[0];
    const int*   ei    = (const int*)  d_in[1];
    const float* w1    = (const float*)d_in[2];
    const float* b1    = (const float*)d_in[3];
    const float* beta2 = (const float*)d_in[4];
    const float* w2    = (const float*)d_in[5];
    const float* b2    = (const float*)d_in[6];

    float* out = (float*)d_out;                 // [N*C] log-softmax ...
    float* x1  = out + (size_t)NN * CC;         // ... then [N*H] x1 (2nd output)

    float* ws  = (float*)d_ws;
    float* h   = ws;                            // N*16
    float* xn  = h  + (size_t)NN * HH;          // N*16 (reused for both convs)
    float* h2  = xn + (size_t)NN * HH;          // N*16
    float* den = h2 + (size_t)NN * HH;          // N
    float* ex  = den + NN;                      // E+N

    const int EP      = EE + NN;
    const int blkNode = (NN + 255) / 256;
    const int blkEdge = (EP + 255) / 256;
    const int blkGemm = ((NN / 16) * 32 + 255) / 256;   // one wave per 16-row tile
    const int blkNH   = (NN * HH + 255) / 256;

    // 1) h = relu(x W1^T + b1)  [WMMA]
    gemm1_relu_kernel<<<blkGemm, 256, 0, stream>>>(x, w1, b1, h);

    // 2) conv1: xn = normalize(h); softmax-weighted aggregation -> x1 (in d_out)
    normalize_kernel<<<blkNode, 256, 0, stream>>>(h, xn);
    zero_kernel<<<blkNode, 256, 0, stream>>>(den, NN);
    zero_kernel<<<blkNH,   256, 0, stream>>>(x1, NN * HH);
    edge_num_kernel<<<blkEdge, 256, 0, stream>>>(ei, xn, nullptr, ex, den);
    edge_agg_kernel<<<blkEdge, 256, 0, stream>>>(ei, ex, den, h, x1);

    // 3) conv2: same with learnable beta2, features = x1 -> h2
    normalize_kernel<<<blkNode, 256, 0, stream>>>(x1, xn);
    zero_kernel<<<blkNode, 256, 0, stream>>>(den, NN);
    zero_kernel<<<blkNH,   256, 0, stream>>>(h2, NN * HH);
    edge_num_kernel<<<blkEdge, 256, 0, stream>>>(ei, xn, beta2, ex, den);
    edge_agg_kernel<<<blkEdge, 256, 0, stream>>>(ei, ex, den, x1, h2);

    // 4) logits + log_softmax -> d_out[0 : N*C]
    head_kernel<<<blkNode, 256, 0, stream>>>(h2, w2, b2, out);
}